// MaskedObjectPooling_63660005261835
// MI455X (gfx1250) — compile-verified
//
#include <hip/hip_runtime.h>
#include <hip/hip_bf16.h>
#include <math.h>

// ---------------- problem constants ----------------
#define BB   8
#define NN   576          // 24*24 patch grid
#define DD   1024
#define KK   64           // objects
#define HH   512
#define WW   512
#define OUT  256
#define HF   24           // sqrt(N)
#define XP   1032         // D + 5 geom, padded to multiple of 4
#define KVALID 1029       // D + 5

typedef __attribute__((ext_vector_type(2))) float v2f;
typedef __attribute__((ext_vector_type(8))) float v8f;

// D = A(16x4 f32) * B(4x16 f32) + C(16x16 f32), full f32 precision.
#define WMMA_F32(a, b, c) \
  __builtin_amdgcn_wmma_f32_16x16x4_f32(false, (a), false, (b), (short)0, (c), false, false)

// Binary-weighted byte-index masks: bool bytes are 0x00/0x01, so set bits of a
// 64-bit word sit exactly at bit positions 8*i.  sum(i * byte_i) =
// popc(w&M1) + 2*popc(w&M2) + 4*popc(w&M4).
#define M1 0x0100010001000100ULL   // byte indices {1,3,5,7}
#define M2 0x0101000001010000ULL   // byte indices {2,3,6,7}
#define M4 0x0101010100000000ULL   // byte indices {4,5,6,7}

// ---------------------------------------------------------------------------
// Kernel 1: per-(b,k) mask statistics + 24x24 nearest downsample + geometry.
// One block per (b,k).  Popcount-based word processing (~1.2 VALU ops/byte)
// keeps this kernel HBM-bound (128 MB of masks ~ 5.5 us at 23.3 TB/s).
// ---------------------------------------------------------------------------
__global__ void mask_stats_kernel(const unsigned char* __restrict__ masks,
                                  float* __restrict__ m_small,
                                  float* __restrict__ inv_msum,
                                  float* __restrict__ x_ws) {
  const int bk  = blockIdx.x;        // 0..511
  const int tid = threadIdx.x;       // 0..255
  const unsigned char* m = masks + (size_t)bk * (HH * WW);

  __shared__ int   s_area, s_sumx, s_sumy, s_minx, s_maxx, s_miny, s_maxy;
  __shared__ float s_msum;
  if (tid == 0) {
    s_area = 0; s_sumx = 0; s_sumy = 0;
    s_minx = 1 << 30; s_maxx = -1; s_miny = 1 << 30; s_maxy = -1;
    s_msum = 0.0f;
  }
  __syncthreads();

  int area = 0, sumx = 0, sumy = 0;
  int minx = 1 << 30, maxx = -1, miny = 1 << 30, maxy = -1;

  // 512 rows / 256 threads = 2 rows per thread; 512B per row as 32 x uint4.
  for (int rr = 0; rr < 2; ++rr) {
    const int y = tid * 2 + rr;
    const uint4* row4 = (const uint4*)(m + (size_t)y * WW);
    int ra = 0, rsx = 0, rminx = 1 << 30, rmaxx = -1;
    for (int c16 = 0; c16 < WW / 16; ++c16) {
      const uint4 v = row4[c16];
      unsigned long long wlo = (unsigned long long)v.x |
                               ((unsigned long long)v.y << 32);
      unsigned long long whi = (unsigned long long)v.z |
                               ((unsigned long long)v.w << 32);
      const int xb = c16 * 16;
#pragma unroll
      for (int half = 0; half < 2; ++half) {
        const unsigned long long w = half ? whi : wlo;
        if (w) {
          const int base = xb + half * 8;
          const int cnt  = __popcll(w);
          const int inner = __popcll(w & M1) + (__popcll(w & M2) << 1) +
                            (__popcll(w & M4) << 2);
          ra  += cnt;
          rsx += cnt * base + inner;
          const int lo = base + ((__ffsll((long long)w) - 1) >> 3);
          const int hi = base + ((63 - __clzll((long long)w)) >> 3);
          rminx = min(rminx, lo);
          rmaxx = max(rmaxx, hi);
        }
      }
    }
    if (ra > 0) {
      area += ra; sumx += rsx; sumy += y * ra;
      minx = min(minx, rminx); maxx = max(maxx, rmaxx);
      miny = min(miny, y);     maxy = max(maxy, y);
    }
  }
  atomicAdd(&s_area, area); atomicAdd(&s_sumx, sumx); atomicAdd(&s_sumy, sumy);
  atomicMin(&s_minx, minx); atomicMax(&s_maxx, maxx);
  atomicMin(&s_miny, miny); atomicMax(&s_maxy, maxy);

  // nearest-neighbor downsample to 24x24 patch grid
  float local = 0.0f;
  for (int idx = tid; idx < NN; idx += 256) {
    const int i = idx / HF, j = idx % HF;
    const int r = (i * HH) / HF, c = (j * WW) / HF;
    const float v = m[(size_t)r * WW + c] ? 1.0f : 0.0f;
    m_small[(size_t)bk * NN + idx] = v;
    local += v;
  }
  atomicAdd(&s_msum, local);
  __syncthreads();

  if (tid == 0) {
    inv_msum[bk] = 1.0f / fmaxf(s_msum, 1e-6f);
    float g0 = 0.f, g1 = 0.f, g2 = 0.f, g3 = 0.f, g4 = 0.f;
    if (s_area >= 1) {
      const float areaf = (float)s_area;
      const float safe  = fmaxf(areaf, 1.0f);
      g0 = ((float)s_sumx / safe) / (float)WW;          // cx
      g1 = ((float)s_sumy / safe) / (float)HH;          // cy
      g2 = areaf / (float)(HH * WW);                    // area_norm
      g3 = (float)(s_maxx - s_minx + 1) / (float)WW;    // bbox_w
      g4 = (float)(s_maxy - s_miny + 1) / (float)HH;    // bbox_h
    }
    float* xr = x_ws + (size_t)bk * XP + DD;
    xr[0] = g0; xr[1] = g1; xr[2] = g2; xr[3] = g3; xr[4] = g4;
    xr[5] = 0.f; xr[6] = 0.f; xr[7] = 0.f;  // K-pad 1029..1031
  }
}

// ---------------------------------------------------------------------------
// Kernel 2: background mean over N per (b, d).
// ---------------------------------------------------------------------------
__global__ void bg_mean_kernel(const float* __restrict__ F,
                               float* __restrict__ mean) {
  const int b = blockIdx.x >> 2;
  const int d = ((blockIdx.x & 3) << 8) + threadIdx.x;
  const float* p = F + (size_t)b * NN * DD + d;
  float s = 0.0f;
  for (int n = 0; n < NN; ++n) s += p[(size_t)n * DD];
  mean[b * DD + d] = s * (1.0f / (float)NN);
}

// ---------------------------------------------------------------------------
// Kernel 3: pooled = (m_small @ F) * inv_msum - mean, via f32 WMMA.
// Grid: 8 batches x 8 N-chunks of 128; 8 waves/block, one 16-wide N tile each;
// each wave loops over the four 16-row M tiles. K = 576 in steps of 4.
// (F is 18.9 MB -> L2-resident on the 192 MB L2 after bg_mean touches it.)
// ---------------------------------------------------------------------------
__global__ void pool_wmma_kernel(const float* __restrict__ F,
                                 const float* __restrict__ m_small,
                                 const float* __restrict__ inv_msum,
                                 const float* __restrict__ mean,
                                 float* __restrict__ x_ws) {
  const int b     = blockIdx.x >> 3;
  const int nbase = (blockIdx.x & 7) * 128;
  const int wv    = threadIdx.x >> 5;
  const int lane  = threadIdx.x & 31;
  const int n0    = nbase + wv * 16;
  const int lh    = lane >> 4;     // half-wave selects K pair
  const int ll    = lane & 15;
  const float* Fb = F + (size_t)b * NN * DD;

  for (int m0 = 0; m0 < KK; m0 += 16) {
    v8f c = {};
    const float* arow = m_small + (size_t)(b * KK + m0 + ll) * NN + 2 * lh;
    const float* bcol = Fb + (size_t)(2 * lh) * DD + n0 + ll;
    for (int k = 0; k < NN; k += 4) {
      v2f a;  a.x  = arow[k];                 a.y  = arow[k + 1];
      v2f bb; bb.x = bcol[(size_t)k * DD];    bb.y = bcol[(size_t)k * DD + DD];
      c = WMMA_F32(a, bb, c);
    }
#pragma unroll
    for (int r = 0; r < 8; ++r) {
      const int mrow = m0 + r + 8 * lh;
      const int row  = b * KK + mrow;
      const int n    = n0 + ll;
      x_ws[(size_t)row * XP + n] = c[r] * inv_msum[row] - mean[b * DD + n];
    }
  }
}

// ---------------------------------------------------------------------------
// Kernel 4: h = gelu(x @ w1 + b1), exact erf.  M=512, N=256, K=1032 (padded).
// ---------------------------------------------------------------------------
__global__ void fc1_wmma_kernel(const float* __restrict__ x_ws,
                                const float* __restrict__ w1,
                                const float* __restrict__ b1,
                                float* __restrict__ h_ws) {
  const int wv   = threadIdx.x >> 5;
  const int lane = threadIdx.x & 31;
  const int tile = blockIdx.x * 8 + wv;          // 0..511
  const int m0   = (tile >> 4) << 4;             // 32 M tiles
  const int n0   = (tile & 15) << 4;             // 16 N tiles
  const int lh   = lane >> 4, ll = lane & 15;

  v8f c = {};
  const float* arow = x_ws + (size_t)(m0 + ll) * XP + 2 * lh;
  for (int k = 0; k < XP; k += 4) {
    v2f a; a.x = arow[k]; a.y = arow[k + 1];
    const int kk = k + 2 * lh;
    v2f bb;
    bb.x = (kk     < KVALID) ? w1[(size_t)kk       * OUT + n0 + ll] : 0.0f;
    bb.y = (kk + 1 < KVALID) ? w1[(size_t)(kk + 1) * OUT + n0 + ll] : 0.0f;
    c = WMMA_F32(a, bb, c);
  }
#pragma unroll
  for (int r = 0; r < 8; ++r) {
    const int row = m0 + r + 8 * lh;
    const int n   = n0 + ll;
    float v = c[r] + b1[n];
    v = 0.5f * v * (1.0f + erff(v * 0.70710678118654752440f));
    h_ws[(size_t)row * OUT + n] = v;
  }
}

// ---------------------------------------------------------------------------
// Kernel 5: LayerNorm over OUT=256 per row. One block per row.
// ---------------------------------------------------------------------------
__global__ void ln_kernel(const float* __restrict__ h,
                          const float* __restrict__ g,
                          const float* __restrict__ bta,
                          float* __restrict__ hn) {
  __shared__ float red[256];
  const int row = blockIdx.x, t = threadIdx.x;
  const float v = h[(size_t)row * OUT + t];

  red[t] = v; __syncthreads();
  for (int s = 128; s > 0; s >>= 1) { if (t < s) red[t] += red[t + s]; __syncthreads(); }
  const float mu = red[0] * (1.0f / (float)OUT);
  __syncthreads();

  const float d = v - mu;
  red[t] = d * d; __syncthreads();
  for (int s = 128; s > 0; s >>= 1) { if (t < s) red[t] += red[t + s]; __syncthreads(); }
  const float var = red[0] * (1.0f / (float)OUT);

  hn[(size_t)row * OUT + t] = d * rsqrtf(var + 1e-5f) * g[t] + bta[t];
}

// ---------------------------------------------------------------------------
// Kernel 6: out = hn @ w2 + b2.  M=512, N=256, K=256.
// ---------------------------------------------------------------------------
__global__ void fc2_wmma_kernel(const float* __restrict__ hn,
                                const float* __restrict__ w2,
                                const float* __restrict__ b2,
                                float* __restrict__ out) {
  const int wv   = threadIdx.x >> 5;
  const int lane = threadIdx.x & 31;
  const int tile = blockIdx.x * 8 + wv;
  const int m0   = (tile >> 4) << 4;
  const int n0   = (tile & 15) << 4;
  const int lh   = lane >> 4, ll = lane & 15;

  v8f c = {};
  const float* arow = hn + (size_t)(m0 + ll) * OUT + 2 * lh;
  const float* bcol = w2 + (size_t)(2 * lh) * OUT + n0 + ll;
  for (int k = 0; k < OUT; k += 4) {
    v2f a;  a.x  = arow[k];               a.y  = arow[k + 1];
    v2f bb; bb.x = bcol[(size_t)k * OUT]; bb.y = bcol[(size_t)k * OUT + OUT];
    c = WMMA_F32(a, bb, c);
  }
#pragma unroll
  for (int r = 0; r < 8; ++r) {
    const int row = m0 + r + 8 * lh;
    const int n   = n0 + ll;
    out[(size_t)row * OUT + n] = c[r] + b2[n];
  }
}

// ---------------------------------------------------------------------------
extern "C" void kernel_launch(void* const* d_in, const int* in_sizes, int n_in,
                              void* d_out, int out_size, void* d_ws, size_t ws_size,
                              hipStream_t stream) {
  const float*         F     = (const float*)d_in[0];          // [8,576,1024]
  const unsigned char* masks = (const unsigned char*)d_in[1];  // [8,64,512,512] bool
  const float*         w1    = (const float*)d_in[2];          // [1029,256]
  const float*         b1    = (const float*)d_in[3];          // [256]
  const float*         ln_g  = (const float*)d_in[4];          // [256]
  const float*         ln_b  = (const float*)d_in[5];          // [256]
  const float*         w2    = (const float*)d_in[6];          // [256,256]
  const float*         b2    = (const float*)d_in[7];          // [256]
  float*               out   = (float*)d_out;                  // [8,64,256]

  // workspace layout (floats)
  float* ws       = (float*)d_ws;
  float* m_small  = ws;                                 // 512*576
  float* inv_msum = m_small + (size_t)BB * KK * NN;     // 512
  float* mean     = inv_msum + BB * KK;                 // 8*1024
  float* x_ws     = mean + BB * DD;                     // 512*1032
  float* h_ws     = x_ws + (size_t)BB * KK * XP;        // 512*256
  float* hn_ws    = h_ws + (size_t)BB * KK * OUT;       // 512*256

  mask_stats_kernel<<<BB * KK, 256, 0, stream>>>(masks, m_small, inv_msum, x_ws);
  bg_mean_kernel<<<BB * (DD / 256), 256, 0, stream>>>(F, mean);
  pool_wmma_kernel<<<BB * (DD / 128), 256, 0, stream>>>(F, m_small, inv_msum, mean, x_ws);
  fc1_wmma_kernel<<<64, 256, 0, stream>>>(x_ws, w1, b1, h_ws);
  ln_kernel<<<BB * KK, 256, 0, stream>>>(h_ws, ln_g, ln_b, hn_ws);
  fc2_wmma_kernel<<<64, 256, 0, stream>>>(hn_ws, w2, b2, out);
}